// TagCosineLoss_87136296501788
// MI455X (gfx1250) — compile-verified
//
#include <hip/hip_runtime.h>
#include <math.h>

#define BATCH 16
#define NPTS  16384
#define DIM   256
#define NGRP  256
#define EPSL  1e-6f

typedef __attribute__((ext_vector_type(2))) float v2f;
typedef __attribute__((ext_vector_type(8))) float v8f;

__device__ __forceinline__ float wsum(float v) {
#pragma unroll
  for (int o = 16; o > 0; o >>= 1) v += __shfl_xor(v, o, 32);
  return v;
}

// ---------------------------------------------------------------------------
// K1: deterministic segment sum. grid = BATCH*16 blocks, 512 thr (16 waves).
// Wave owns group g = part*16 + wave. Ids staged in LDS; wave ballot-scans
// ids and accumulates only its group's pred rows (8 floats/lane).
// ---------------------------------------------------------------------------
__global__ void __launch_bounds__(512) k_segsum(const float* __restrict__ pred,
                                                const int* __restrict__ gids,
                                                float* __restrict__ sums,
                                                float* __restrict__ counts) {
  extern __shared__ int s_ids[];
  const int b    = blockIdx.x >> 4;
  const int part = blockIdx.x & 15;
  const int tid  = threadIdx.x;
  const int lane = tid & 31;
  const int wave = tid >> 5;
  const int g    = part * 16 + wave;

  const int* gb = gids + b * NPTS;
  for (int i = tid; i < NPTS; i += 512) s_ids[i] = gb[i];
  __syncthreads();

  const int d0 = lane * 8;
  const float* predb = pred + (size_t)b * NPTS * DIM;
  float a0 = 0.f, a1 = 0.f, a2 = 0.f, a3 = 0.f;
  float a4 = 0.f, a5 = 0.f, a6 = 0.f, a7 = 0.f;
  float cnt = 0.0f;

  for (int i0 = 0; i0 < NPTS; i0 += 32) {
    const int gi = s_ids[i0 + lane];
    unsigned m = __builtin_amdgcn_ballot_w32(gi == g);
    cnt += (float)__popc(m);
    while (m) {
      const int j = __ffs((int)m) - 1;
      m &= m - 1;
      const float4* row = (const float4*)(predb + (size_t)(i0 + j) * DIM + d0);
      const float4 p0 = row[0];
      const float4 p1 = row[1];
      a0 += p0.x; a1 += p0.y; a2 += p0.z; a3 += p0.w;
      a4 += p1.x; a5 += p1.y; a6 += p1.z; a7 += p1.w;
    }
  }

  float* srow = sums + ((size_t)b * NGRP + g) * DIM + d0;
  ((float4*)srow)[0] = make_float4(a0, a1, a2, a3);
  ((float4*)srow)[1] = make_float4(a4, a5, a6, a7);
  if (lane == 0) counts[b * NGRP + g] = cnt;
}

// ---------------------------------------------------------------------------
// K2: tags = sums / max(count,1) (in place), tn = ||tag||. grid = BATCH,
// 256 thr (8 waves); each wave handles 32 groups, lane covers 8 dims.
// ---------------------------------------------------------------------------
__global__ void __launch_bounds__(256) k_tags(float* __restrict__ tags,
                                              const float* __restrict__ counts,
                                              float* __restrict__ tn) {
  const int b    = blockIdx.x;
  const int lane = threadIdx.x & 31;
  const int wave = threadIdx.x >> 5;
  const int d0   = lane * 8;
#pragma unroll 1
  for (int t = 0; t < 32; ++t) {
    const int g = wave * 32 + t;
    const float c  = counts[b * NGRP + g];
    const float rc = 1.0f / fmaxf(c, 1.0f);
    float* row = tags + ((size_t)b * NGRP + g) * DIM + d0;
    float4 s0 = ((float4*)row)[0];
    float4 s1 = ((float4*)row)[1];
    s0.x *= rc; s0.y *= rc; s0.z *= rc; s0.w *= rc;
    s1.x *= rc; s1.y *= rc; s1.z *= rc; s1.w *= rc;
    ((float4*)row)[0] = s0;
    ((float4*)row)[1] = s1;
    float n2 = s0.x*s0.x + s0.y*s0.y + s0.z*s0.z + s0.w*s0.w
             + s1.x*s1.x + s1.y*s1.y + s1.z*s1.z + s1.w*s1.w;
    n2 = wsum(n2);
    if (lane == 0) tn[b * NGRP + g] = sqrtf(n2);
  }
}

// ---------------------------------------------------------------------------
// K3: pull term. Same ownership structure as K1; the owning wave holds its
// group's tag row in registers; per member element compute 1 - cos and
// accumulate. pullg = sum / max(count,1). Deterministic, atomic-free.
// ---------------------------------------------------------------------------
__global__ void __launch_bounds__(512) k_pull(const float* __restrict__ pred,
                                              const int* __restrict__ gids,
                                              const float* __restrict__ tags,
                                              const float* __restrict__ counts,
                                              const float* __restrict__ tn,
                                              float* __restrict__ pullg) {
  extern __shared__ int s_ids[];
  const int b    = blockIdx.x >> 4;
  const int part = blockIdx.x & 15;
  const int tid  = threadIdx.x;
  const int lane = tid & 31;
  const int wave = tid >> 5;
  const int g    = part * 16 + wave;

  const int* gb = gids + b * NPTS;
  for (int i = tid; i < NPTS; i += 512) s_ids[i] = gb[i];
  __syncthreads();

  const int d0 = lane * 8;
  const float* trow = tags + ((size_t)b * NGRP + g) * DIM + d0;
  const float4 t0 = ((const float4*)trow)[0];
  const float4 t1 = ((const float4*)trow)[1];
  const float tng = tn[b * NGRP + g];
  const float c   = counts[b * NGRP + g];
  const float* predb = pred + (size_t)b * NPTS * DIM;

  float acc = 0.0f;
  for (int i0 = 0; i0 < NPTS; i0 += 32) {
    const int gi = s_ids[i0 + lane];
    unsigned m = __builtin_amdgcn_ballot_w32(gi == g);
    while (m) {
      const int j = __ffs((int)m) - 1;
      m &= m - 1;
      const float4* row = (const float4*)(predb + (size_t)(i0 + j) * DIM + d0);
      const float4 p0 = row[0];
      const float4 p1 = row[1];
      float dp = p0.x*t0.x + p0.y*t0.y + p0.z*t0.z + p0.w*t0.w
               + p1.x*t1.x + p1.y*t1.y + p1.z*t1.z + p1.w*t1.w;
      float pp = p0.x*p0.x + p0.y*p0.y + p0.z*p0.z + p0.w*p0.w
               + p1.x*p1.x + p1.y*p1.y + p1.z*p1.z + p1.w*p1.w;
      dp = wsum(dp);
      pp = wsum(pp);
      acc += 1.0f - dp / fmaxf(sqrtf(pp) * tng, EPSL);
    }
  }
  if (lane == 0) pullg[b * NGRP + g] = acc / fmaxf(c, 1.0f);
}

// ---------------------------------------------------------------------------
// K4: push term via f32 WMMA gram of tags, plus final per-image reduction.
// grid = BATCH, 256 thr (8 waves). Tags (256KB) + tn/present staged in LDS;
// each wave computes 32 16x16 tiles of tags@tags^T with V_WMMA_F32_16X16X4_F32
// and applies the exact eps-clamped normalization + presence mask per entry.
// ---------------------------------------------------------------------------
__global__ void __launch_bounds__(256) k_push(const float* __restrict__ tags,
                                              const float* __restrict__ counts,
                                              const float* __restrict__ tn,
                                              const float* __restrict__ pullg,
                                              float* __restrict__ losses) {
  extern __shared__ float s_f[];
  float* s_tags = s_f;                  // NGRP*DIM
  float* s_tn   = s_f + NGRP * DIM;     // NGRP
  float* s_pres = s_tn + NGRP;          // NGRP
  float* s_red  = s_pres + NGRP;        // 64 scratch

  const int b    = blockIdx.x;
  const int tid  = threadIdx.x;
  const int lane = tid & 31;
  const int wave = tid >> 5;

  // Stage tags into LDS (coalesced float4).
  const float4* tg = (const float4*)(tags + (size_t)b * NGRP * DIM);
  float4* st = (float4*)s_tags;
  for (int i = tid; i < NGRP * DIM / 4; i += 256) st[i] = tg[i];

  // One thread per group: presence, norms, pull partials.
  const float cg   = counts[b * NGRP + tid];
  const float pres = cg > 0.0f ? 1.0f : 0.0f;
  s_tn[tid]   = tn[b * NGRP + tid];
  s_pres[tid] = pres;
  float on = wsum(pres);
  float pl = wsum(pres * pullg[b * NGRP + tid]);
  if (lane == 0) { s_red[wave] = on; s_red[8 + wave] = pl; }
  __syncthreads();
  if (tid == 0) {
    float o = 0.f, p = 0.f;
    for (int w = 0; w < 8; ++w) { o += s_red[w]; p += s_red[8 + w]; }
    s_red[16] = o; s_red[17] = p;
  }
  __syncthreads();
  const float objn = s_red[16];

  // WMMA tile loop: A layout (16x4 f32): lane l holds A[l&15][2*(l>>4)+{0,1}];
  // B (4x16) mirrors it, so for B = tags^T both operands gather tag rows.
  const int l15  = lane & 15;
  const int rsel = lane >> 4;
  float partial = 0.0f;
  for (int t = wave; t < 256; t += 8) {
    const int R  = t >> 4;
    const int Cb = t & 15;
    const float* arow = s_tags + (R * 16 + l15) * DIM + 2 * rsel;
    const float* brow = s_tags + (Cb * 16 + l15) * DIM + 2 * rsel;
    v8f cacc = {0.f, 0.f, 0.f, 0.f, 0.f, 0.f, 0.f, 0.f};
#pragma unroll 4
    for (int k0 = 0; k0 < DIM; k0 += 4) {
      const v2f av = *(const v2f*)(arow + k0);
      const v2f bv = *(const v2f*)(brow + k0);
      cacc = __builtin_amdgcn_wmma_f32_16x16x4_f32(
          false, av, false, bv, (short)0, cacc, false, false);
    }
    // C/D layout: vgpr r -> M = R*16 + r + 8*(lane>=16); N = Cb*16 + (lane&15)
    const float pb  = s_pres[Cb * 16 + l15];
    const float tnb = s_tn[Cb * 16 + l15];
#pragma unroll
    for (int r = 0; r < 8; ++r) {
      const int aRow = R * 16 + r + 8 * rsel;
      const float pa    = s_pres[aRow];
      const float tna   = s_tn[aRow];
      const float denom = fmaxf(tna * tnb, EPSL);
      partial += pa * pb * (1.0f + cacc[r] / denom);
    }
  }
  partial = wsum(partial);
  if (lane == 0) s_red[24 + wave] = partial;
  __syncthreads();
  if (tid == 0) {
    float sumC = 0.f;
    for (int w = 0; w < 8; ++w) sumC += s_red[24 + w];
    const float pullsum = s_red[17];
    const float pull = pullsum / (objn + EPSL);
    const float push = (sumC - 2.0f * objn) / (((objn - 1.0f) * objn + EPSL) * 2.0f);
    losses[b] = (objn > 1.0f) ? (push + pull) : 0.0f;
  }
}

// ---------------------------------------------------------------------------
// K5: mean over images -> scalar output (deterministic single block).
// ---------------------------------------------------------------------------
__global__ void k_final(const float* __restrict__ losses, float* __restrict__ out) {
  float v = (threadIdx.x < BATCH) ? losses[threadIdx.x] : 0.0f;
  v = wsum(v);
  if (threadIdx.x == 0) out[0] = v * (1.0f / (float)BATCH);
}

extern "C" void kernel_launch(void* const* d_in, const int* in_sizes, int n_in,
                              void* d_out, int out_size, void* d_ws, size_t ws_size,
                              hipStream_t stream) {
  (void)in_sizes; (void)n_in; (void)out_size; (void)ws_size;
  const float* pred = (const float*)d_in[0];
  const int*   gids = (const int*)d_in[1];
  // d_in[2] (anchor_inds) is unused by the reference loss.

  float* ws     = (float*)d_ws;
  float* sums   = ws;                                   // [B][G][D], becomes tags
  float* counts = sums + (size_t)BATCH * NGRP * DIM;    // [B][G]
  float* tn     = counts + BATCH * NGRP;                // [B][G]
  float* pullg  = tn + BATCH * NGRP;                    // [B][G]
  float* losses = pullg + BATCH * NGRP;                 // [B]

  const size_t ids_lds  = (size_t)NPTS * sizeof(int);                       // 64 KB
  const size_t push_lds = (size_t)(NGRP * DIM + 2 * NGRP + 64) * sizeof(float); // ~258 KB

  k_segsum<<<dim3(BATCH * 16), dim3(512), ids_lds, stream>>>(pred, gids, sums, counts);
  k_tags  <<<dim3(BATCH),      dim3(256), 0,       stream>>>(sums, counts, tn);
  k_pull  <<<dim3(BATCH * 16), dim3(512), ids_lds, stream>>>(pred, gids, sums, counts, tn, pullg);
  k_push  <<<dim3(BATCH),      dim3(256), push_lds, stream>>>(sums, counts, tn, pullg, losses);
  k_final <<<dim3(1),          dim3(32),  0,       stream>>>(losses, (float*)d_out);
}